// BidirectionalTrustModel_27874337751766
// MI455X (gfx1250) — compile-verified
//
#include <hip/hip_runtime.h>

#define NBINS 10

// CAP_MATRIX.T, padded to float4 per row. Values folded in fp32 exactly as
// the reference computes them (0.01f * v) — matters for the step<=edge
// boundary comparisons (e.g. step 0.45 vs cap 0.45).
__constant__ float CAPT[13][4] = {
    {0.01f *  0.0f, 0.01f *  0.0f, 0.01f *  0.0f, 0.0f},
    {0.01f * 33.0f, 0.01f * 33.0f, 0.01f * 33.0f, 0.0f},
    {0.01f * 50.0f, 0.01f * 49.0f, 0.01f * 42.0f, 0.0f},
    {0.01f * 43.0f, 0.01f * 39.0f, 0.01f * 39.0f, 0.0f},
    {0.01f * 56.0f, 0.01f * 58.0f, 0.01f * 44.0f, 0.0f},
    {0.01f * 67.0f, 0.01f * 67.0f, 0.01f * 52.0f, 0.0f},
    {0.01f * 62.0f, 0.01f * 60.0f, 0.01f * 49.0f, 0.0f},
    {0.01f * 47.0f, 0.01f * 54.0f, 0.01f * 42.0f, 0.0f},
    {0.01f * 50.0f, 0.01f * 52.0f, 0.01f * 45.0f, 0.0f},
    {0.01f * 51.0f, 0.01f * 52.0f, 0.01f * 46.0f, 0.0f},
    {0.01f * 64.0f, 0.01f * 67.0f, 0.01f * 52.0f, 0.0f},
    {0.01f * 64.0f, 0.01f * 69.0f, 0.01f * 53.0f, 0.0f},
    {0.01f * 68.0f, 0.01f * 71.0f, 0.01f * 56.0f, 0.0f},
};

__global__ __launch_bounds__(256) void bidir_trust_kernel(
    const int2*  __restrict__ perf,     // [T][N] packed (p0,p1)
    const int*   __restrict__ obsids,   // [T][N]
    const int*   __restrict__ predids,  // [N]
    const float* __restrict__ betas,    // [3]
    const float* __restrict__ zetas,    // [3]
    float*       __restrict__ out,      // [N]
    int T, int N)
{
    // caps[id]          : CAP_MATRIX.T row (float4-padded) for the scan loop
    // PS[id][c][k]      : prefix sum over bins b<k of
    //                     d = (1+exp(beta_c*(req_{id,c}-step_b)))^(-zeta_c^2)
    //                     (same accumulation order as the reference loop)
    __shared__ float4 caps[13];
    __shared__ float  PS[13][3][NBINS + 1];

    const int tid = threadIdx.x;
    if (tid < 13) {
        caps[tid] = ((const float4*)CAPT)[tid];
    }
    if (tid < 39) {  // 13 ids x 3 dims: only 390 distinct d-values exist
        const int   id   = tid / 3;
        const int   c    = tid % 3;
        const float req  = CAPT[id][c];
        const float beta = betas[c];
        const float zz   = zetas[c] * zetas[c];
        float acc = 0.0f;
        PS[id][c][0] = 0.0f;
#pragma unroll
        for (int b = 0; b < NBINS; ++b) {
            const float step = ((float)b + 0.5f) / 10.0f;  // exact division
            acc += powf(1.0f + expf(beta * (req - step)), -zz);
            PS[id][c][b + 1] = acc;
        }
    }
    __syncthreads();

    const int n = blockIdx.x * blockDim.x + tid;
    if (n >= N) return;

    // ---- sequential scan over T timesteps (min/max clamp updates) ----
    float e0 = 0.0f, e1 = 0.0f, e2 = 0.0f;
    const int PF = 8;  // prefetch distance in t-slices
#pragma unroll 4
    for (int t = 0; t < T; ++t) {
        const size_t base = (size_t)t * (size_t)N + (size_t)n;
        if (t + PF < T) {
            // gfx1250 global_prefetch_b8: hide the large inter-slice stride
            __builtin_prefetch(&perf[base + (size_t)PF * N], 0, 3);
            __builtin_prefetch(&obsids[base + (size_t)PF * N], 0, 3);
        }
        const int2   p  = perf[base];     // coalesced 8B/lane
        const int    id = obsids[base];   // coalesced 4B/lane
        const float4 c  = caps[id];       // ds_load_b128, conflict-free

        const bool nonzero = (p.x + p.y) > 0;
        const bool success = (p.y > 0) && (p.x == 0);
        const bool up   = nonzero && success;
        const bool down = nonzero && !success;
        e0 = up ? fmaxf(e0, c.x) : (down ? fminf(e0, c.x) : e0);
        e1 = up ? fmaxf(e1, c.y) : (down ? fminf(e1, c.y) : e1);
        e2 = up ? fmaxf(e2, c.z) : (down ? fminf(e2, c.z) : e2);
    }

    // ---- closed-form collapse of the 10x10x10 einsum ----
    // trust = (D0*D1*D2 - Z0*Z1*Z2) / (1000 - c0*c1*c2)
    // with D_c = PS[pid][c][10], Z_c = PS[pid][c][cnt_c],
    // cnt_c = #{ step_b <= edge_c }  (identical float comparisons as JAX)
    int c0 = 0, c1 = 0, c2 = 0;
#pragma unroll
    for (int b = 0; b < NBINS; ++b) {
        const float step = ((float)b + 0.5f) / 10.0f;
        c0 += (step <= e0);
        c1 += (step <= e1);
        c2 += (step <= e2);
    }

    const int pid = predids[n];
    const float D0 = PS[pid][0][NBINS];
    const float D1 = PS[pid][1][NBINS];
    const float D2 = PS[pid][2][NBINS];
    const float Z0 = PS[pid][0][c0];
    const float Z1 = PS[pid][1][c1];
    const float Z2 = PS[pid][2][c2];

    const float S = 1000.0f - (float)(c0 * c1 * c2);  // >= 657, exact in fp32
    out[n] = (D0 * D1 * D2 - Z0 * Z1 * Z2) / S;
}

extern "C" void kernel_launch(void* const* d_in, const int* in_sizes, int n_in,
                              void* d_out, int out_size, void* d_ws, size_t ws_size,
                              hipStream_t stream) {
    // setup_inputs order:
    // 0: inptasksobs (unused)   1: inptasksperf [T,N,2] i32
    // 2: inptaskspred (unused)  3: num_obs_tasks (scalar, unused host-side)
    // 4: tasksobsids [T,N,1]    5: taskspredids [N,1]
    // 6: betas [3] f32          7: zetas [3] f32
    const int2*  perf    = (const int2*)d_in[1];
    const int*   obsids  = (const int*)d_in[4];
    const int*   predids = (const int*)d_in[5];
    const float* betas   = (const float*)d_in[6];
    const float* zetas   = (const float*)d_in[7];
    float* out = (float*)d_out;

    const int N = out_size;            // 16384
    const int T = in_sizes[4] / N;     // tasksobsids has T*N elements

    const int block = 256;             // 8 wave32 waves per workgroup
    const int grid  = (N + block - 1) / block;
    bidir_trust_kernel<<<grid, block, 0, stream>>>(perf, obsids, predids,
                                                   betas, zetas, out, T, N);
}